// ExpHydroM100_59390807769671
// MI455X (gfx1250) — compile-verified
//
#include <hip/hip_runtime.h>

#define HID 32

typedef float v2f __attribute__((ext_vector_type(2)));
typedef float v8f __attribute__((ext_vector_type(8)));

// ---------- fast scalar math (CDNA5 transcendental units) ----------
static __device__ __forceinline__ float fast_rcp(float x) {
    return __builtin_amdgcn_rcpf(x);
}

static __device__ __forceinline__ float tanh_fast(float x) {
#if __has_builtin(__builtin_amdgcn_tanhf)
    return __builtin_amdgcn_tanhf(x);          // v_tanh_f32
#else
    float e = __expf(2.0f * x);                // v_exp_f32
    return 1.0f - 2.0f * fast_rcp(e + 1.0f);   // v_rcp_f32; saturates to +-1
#endif
}

static __device__ __forceinline__ float sinh_fast(float x) {
    float e = __expf(x);
    return 0.5f * (e - fast_rcp(e));
}

static __device__ __forceinline__ float step_fn(float x) {
    return 0.5f * (tanh_fast(5.0f * x) + 1.0f);
}

static __device__ __forceinline__ float relu(float x) { return fmaxf(x, 0.0f); }

// =====================================================================
// Kernel 1: serial RK4 scan, single wave32.
// Lane j owns hidden unit j (W1/W2 column j, W3 row j).
// Gather for step i+1 and time[i+3] are issued before the (long)
// dependent computation of step i, hiding memory latency.
// =====================================================================
__global__ __launch_bounds__(32) void exphydro_scan(
    const float* __restrict__ inp,    // T x 5 : s_snow,s_water,precp,tmean,time
    const float* __restrict__ lday,   // T
    const float* __restrict__ W1, const float* __restrict__ b1,
    const float* __restrict__ W2, const float* __restrict__ b2,
    const float* __restrict__ W3, const float* __restrict__ b3,
    float2* __restrict__ ys, int T)
{
    const int j = threadIdx.x & 31;
    const int n = T;

    // ---- per-lane weights ----
    float w10 = W1[0 * HID + j], w11 = W1[1 * HID + j];
    float w12 = W1[2 * HID + j], w13 = W1[3 * HID + j];
    float b1j = b1[j];
    float w2j[HID];
#pragma unroll
    for (int k = 0; k < HID; ++k) w2j[k] = W2[k * HID + j];
    float b2j = b2[j];
    float w30 = W3[j * 5 + 0], w31 = W3[j * 5 + 1], w32 = W3[j * 5 + 2];
    float w33 = W3[j * 5 + 3], w34 = W3[j * 5 + 4];
    float b30 = b3[0], b31 = b3[1], b32 = b3[2], b33 = b3[3], b34 = b3[4];

    float y0 = inp[0], y1 = inp[1];
    if (j == 0) ys[0] = make_float2(y0, y1);
    if (T < 2) return;

    float t0 = inp[4];
    float t1 = inp[1 * 5 + 4];
    float t2 = inp[((2 < T - 1) ? 2 : (T - 1)) * 5 + 4];

    // ---- gather lane roles: 3 t-points x {precp,precp+1,tmean,tmean+1,lday,lday+1}
    const int  ll      = j % 18;
    const int  p       = ll / 6;
    const int  sel     = ll % 6;
    const int  pick_ip = sel & 1;
    const int  col     = 2 + (sel >> 1);
    const bool is_ld   = (sel >= 4);
    const float* gbase = is_ld ? lday : (inp + col);
    const int   gstr   = is_ld ? 1 : 5;

    auto calcIdx = [&](float t, int& i0, int& ip, float& fr) {
        float fl = floorf(t);
        fr = t - fl;
        int ii = (int)fl;
        ii = ii % n; if (ii < 0) ii += n;
        i0 = ii;
        ip = ii + 1; if (ip >= n) ip -= n;
    };
    auto gatherOne = [&](int i0a, int ipa, int i0b, int ipb, int i0c, int ipc) -> float {
        int i0p = (p == 0) ? i0a : ((p == 1) ? i0b : i0c);
        int ipp = (p == 0) ? ipa : ((p == 1) ? ipb : ipc);
        int idx = pick_ip ? ipp : i0p;
        return gbase[idx * gstr];
    };

    auto rhs = [&](float s0, float s1, float P, float TM, float LD,
                   float& ka, float& kb) {
        // layer 1: per-lane matvec (lane j = hidden unit j)
        float h = tanh_fast(fmaf(s0, w10, fmaf(s1, w11, fmaf(P, w12, fmaf(TM, w13, b1j)))));
        // layer 2: broadcast h_k, 4 independent accumulator chains
        float a0 = b2j, a1 = 0.f, a2 = 0.f, a3 = 0.f;
#pragma unroll
        for (int k = 0; k < HID; k += 4) {
            a0 = fmaf(__shfl(h, k + 0), w2j[k + 0], a0);
            a1 = fmaf(__shfl(h, k + 1), w2j[k + 1], a1);
            a2 = fmaf(__shfl(h, k + 2), w2j[k + 2], a2);
            a3 = fmaf(__shfl(h, k + 3), w2j[k + 3], a3);
        }
        float h2 = tanh_fast((a0 + a1) + (a2 + a3));
        // layer 3: 5 butterfly reductions (independent chains)
        float o0 = h2 * w30, o1 = h2 * w31, o2 = h2 * w32, o3 = h2 * w33, o4 = h2 * w34;
#pragma unroll
        for (int off = 16; off >= 1; off >>= 1) {
            o0 += __shfl_xor(o0, off);
            o1 += __shfl_xor(o1, off);
            o2 += __shfl_xor(o2, off);
            o3 += __shfl_xor(o3, off);
            o4 += __shfl_xor(o4, off);
        }
        o0 += b30; o1 += b31; o2 += b32; o3 += b33; o4 += b34;

        float psnow = relu(sinh_fast(o0) * step_fn(-TM));
        float prain = relu(sinh_fast(o1));
        float melt  = relu(step_fn(s0) * sinh_fast(o2));
        float s1s   = step_fn(s1);
        float et    = s1s * __expf(o3) * LD;
        float q     = s1s * __expf(o4);
        ka = psnow - melt;
        kb = prain + melt - et - q;
    };

    // ---- prologue gather for step 0 ----
    int i0a, ipa, i0b, ipb, i0c, ipc;
    float fra, frb, frc;
    {
        float dt0 = t1 - t0;
        calcIdx(t0, i0a, ipa, fra);
        calcIdx(t0 + 0.5f * dt0, i0b, ipb, frb);
        calcIdx(t0 + dt0, i0c, ipc, frc);
    }
    float g = gatherOne(i0a, ipa, i0b, ipb, i0c, ipc);

    for (int i = 0; i < T - 1; ++i) {
        // interpolated drivers for current step (g was prefetched last iter)
        float Pa = __shfl(g, 0)  * (1.f - fra) + __shfl(g, 1)  * fra;
        float Ta = __shfl(g, 2)  * (1.f - fra) + __shfl(g, 3)  * fra;
        float La = __shfl(g, 4)  * (1.f - fra) + __shfl(g, 5)  * fra;
        float Pb = __shfl(g, 6)  * (1.f - frb) + __shfl(g, 7)  * frb;
        float Tb = __shfl(g, 8)  * (1.f - frb) + __shfl(g, 9)  * frb;
        float Lb = __shfl(g, 10) * (1.f - frb) + __shfl(g, 11) * frb;
        float Pc = __shfl(g, 12) * (1.f - frc) + __shfl(g, 13) * frc;
        float Tc = __shfl(g, 14) * (1.f - frc) + __shfl(g, 15) * frc;
        float Lc = __shfl(g, 16) * (1.f - frc) + __shfl(g, 17) * frc;

        // prefetch time 3 steps ahead (consumed next iteration)
        int i3 = i + 3; if (i3 > T - 1) i3 = T - 1;
        float t3 = inp[i3 * 5 + 4];

        // issue next step's gather (addresses from t1,t2 already in regs)
        float frna, frnb, frnc;
        int ni0a, nipa, ni0b, nipb, ni0c, nipc;
        {
            float dtn = t2 - t1;
            calcIdx(t1, ni0a, nipa, frna);
            calcIdx(t1 + 0.5f * dtn, ni0b, nipb, frnb);
            calcIdx(t1 + dtn, ni0c, nipc, frnc);
        }
        float gn = gatherOne(ni0a, nipa, ni0b, nipb, ni0c, nipc);

        // RK4
        float dt = t1 - t0;
        float k10, k11, k20, k21, k30, k31, k40, k41;
        rhs(y0, y1, Pa, Ta, La, k10, k11);
        rhs(fmaf(0.5f * dt, k10, y0), fmaf(0.5f * dt, k11, y1), Pb, Tb, Lb, k20, k21);
        rhs(fmaf(0.5f * dt, k20, y0), fmaf(0.5f * dt, k21, y1), Pb, Tb, Lb, k30, k31);
        rhs(fmaf(dt, k30, y0), fmaf(dt, k31, y1), Pc, Tc, Lc, k40, k41);

        float c6 = dt * (1.0f / 6.0f);
        y0 += c6 * (k10 + 2.0f * (k20 + k30) + k40);
        y1 += c6 * (k11 + 2.0f * (k21 + k31) + k41);
        if (j == 0) ys[i + 1] = make_float2(y0, y1);

        // rotate pipeline state
        g = gn; fra = frna; frb = frnb; frc = frnc;
        t0 = t1; t1 = t2; t2 = t3;
    }
}

// =====================================================================
// Kernel 2: batch MLP over all T rows using V_WMMA_F32_16X16X4_F32.
// One wave per 16-row tile. K=32 layers = 8 chained K=4 WMMAs.
// LDS (16 x 32, stride 34) repacks C-layout -> A-layout between layers.
// =====================================================================
#define HSTR 34

static __device__ __forceinline__ v8f wmma4(v2f a, v2f b, v8f c) {
    return __builtin_amdgcn_wmma_f32_16x16x4_f32(false, a, false, b,
                                                 (short)0, c, false, false);
}

__global__ __launch_bounds__(32) void exphydro_mlp_wmma(
    const float* __restrict__ inp,      // T x 5
    const float2* __restrict__ ys,      // T (raw states from scan)
    const float* __restrict__ W1, const float* __restrict__ b1,
    const float* __restrict__ W2, const float* __restrict__ b2,
    const float* __restrict__ W3, const float* __restrict__ b3,
    float* __restrict__ out, int T)
{
    __shared__ float Hs[16 * HSTR];

    const int l  = threadIdx.x & 31;
    const int lo = l & 15;
    const int hi = l >> 4;
    const int kofs = 2 * hi;   // B-layout: lanes 16..31 hold K rows k+2,k+3

    // ---- weights in B layout (v2f per K-slice per N-half) ----
    v2f bw1[2], bw2[8][2], bw3[8];
#pragma unroll
    for (int h = 0; h < 2; ++h) {
        int ncol = h * 16 + lo;
        bw1[h].x = W1[(kofs + 0) * HID + ncol];
        bw1[h].y = W1[(kofs + 1) * HID + ncol];
    }
#pragma unroll
    for (int kk = 0; kk < 8; ++kk) {
        int kr = kk * 4 + kofs;
#pragma unroll
        for (int h = 0; h < 2; ++h) {
            int ncol = h * 16 + lo;
            bw2[kk][h].x = W2[(kr + 0) * HID + ncol];
            bw2[kk][h].y = W2[(kr + 1) * HID + ncol];
        }
        float wa = 0.f, wb = 0.f;
        if (lo < 5) { wa = W3[(kr + 0) * 5 + lo]; wb = W3[(kr + 1) * 5 + lo]; }
        bw3[kk].x = wa; bw3[kk].y = wb;
    }
    float bb1a = b1[lo], bb1b = b1[16 + lo];
    float bb2a = b2[lo], bb2b = b2[16 + lo];
    float bout = b3[4];

    // ---- A tile for layer 1 (16x4 f32 = v2f per lane, direct from memory) ----
    const int r0 = blockIdx.x * 16;
    int r = r0 + lo; if (r > T - 1) r = T - 1;
    float2 yy = ys[r];
    float s0 = fmaxf(yy.x, 0.0f), s1 = fmaxf(yy.y, 0.0f);
    float pc = inp[r * 5 + 2], tm = inp[r * 5 + 3];
    v2f a1;
    a1.x = hi ? pc : s0;   // lanes 0-15: K=0,1 ; lanes 16-31: K=2,3
    a1.y = hi ? tm : s1;

    const v8f vzero = {0.f, 0.f, 0.f, 0.f, 0.f, 0.f, 0.f, 0.f};

    // ---- layer 1: X(16x4) @ W1(4x32) ----
    v8f c0 = vzero, c1 = vzero;
    c0 = wmma4(a1, bw1[0], c0);
    c1 = wmma4(a1, bw1[1], c1);

#pragma unroll
    for (int v = 0; v < 8; ++v) {
        int row = v + 8 * hi;
        Hs[row * HSTR + lo]      = tanh_fast(c0[v] + bb1a);
        Hs[row * HSTR + 16 + lo] = tanh_fast(c1[v] + bb1b);
    }
    __syncthreads();

    // ---- layer 2: H(16x32) @ W2(32x32), 8 chained K=4 WMMAs per N-half ----
    v8f d0 = vzero, d1 = vzero;
#pragma unroll
    for (int kk = 0; kk < 8; ++kk) {
        int col = kk * 4 + kofs;
        v2f a2;
        a2.x = Hs[lo * HSTR + col];
        a2.y = Hs[lo * HSTR + col + 1];
        d0 = wmma4(a2, bw2[kk][0], d0);
        d1 = wmma4(a2, bw2[kk][1], d1);
    }
    __syncthreads();
#pragma unroll
    for (int v = 0; v < 8; ++v) {
        int row = v + 8 * hi;
        Hs[row * HSTR + lo]      = tanh_fast(d0[v] + bb2a);
        Hs[row * HSTR + 16 + lo] = tanh_fast(d1[v] + bb2b);
    }
    __syncthreads();

    // ---- layer 3: H2(16x32) @ W3(32x5 padded) ; only column 4 is needed ----
    v8f e0 = vzero;
#pragma unroll
    for (int kk = 0; kk < 8; ++kk) {
        int col = kk * 4 + kofs;
        v2f a3;
        a3.x = Hs[lo * HSTR + col];
        a3.y = Hs[lo * HSTR + col + 1];
        e0 = wmma4(a3, bw3[kk], e0);
    }
    if (lo == 4) {
#pragma unroll
        for (int v = 0; v < 8; ++v) {
            int row = r0 + v + 8 * hi;
            if (row < T) out[row] = e0[v] + bout;
        }
    }
}

// =====================================================================
extern "C" void kernel_launch(void* const* d_in, const int* in_sizes, int n_in,
                              void* d_out, int out_size, void* d_ws, size_t ws_size,
                              hipStream_t stream) {
    const float* inputs = (const float*)d_in[0];  // T x 5
    const float* lday   = (const float*)d_in[1];  // T
    const float* W1     = (const float*)d_in[2];
    const float* b1     = (const float*)d_in[3];
    const float* W2     = (const float*)d_in[4];
    const float* b2     = (const float*)d_in[5];
    const float* W3     = (const float*)d_in[6];
    const float* b3     = (const float*)d_in[7];
    float*       out    = (float*)d_out;
    const int    T      = in_sizes[1];            // lday length

    float2* ys = (float2*)d_ws;                   // T * 8 bytes of scratch

    exphydro_scan<<<1, 32, 0, stream>>>(inputs, lday, W1, b1, W2, b2, W3, b3, ys, T);

    int tiles = (T + 15) / 16;
    exphydro_mlp_wmma<<<tiles, 32, 0, stream>>>(inputs, ys, W1, b1, W2, b2, W3, b3, out, T);
}